// NextTaskGAT_13821204758819
// MI455X (gfx1250) — compile-verified
//
#include <hip/hip_runtime.h>

// ---------------------------------------------------------------------------
// GAT 2-layer + mean-pool + FC for MI455X (gfx1250, wave32)
// GEMMs use V_WMMA_F32_16X16X4_F32 (fp32 tensor path, matches fp32 reference).
// Edge softmax/aggregation uses one wave32 per edge with f32 global atomics;
// the ~100 MB working set lives in the 192 MB L2, so the irregular scatter
// (~1.7 GB of gather+atomic traffic) stays on-die.
// ---------------------------------------------------------------------------

typedef __attribute__((ext_vector_type(2))) float v2f;
typedef __attribute__((ext_vector_type(8))) float v8f;

#define GAT_N      50000
#define GAT_E      800000LL
#define GAT_TOT    850000LL   // E + N self loops
#define GAT_FIN    128
#define GAT_D1     256        // HEADS*HID
#define GAT_HID    64
#define GAT_HEADS  4
#define GAT_OUT    64
#define GAT_NG     512
#define NEG_SLOPE  0.2f

// -------------------------------- utility ----------------------------------
__global__ void k_fill(float* __restrict__ p, float v, int n) {
  int i = blockIdx.x * blockDim.x + threadIdx.x;
  if (i < n) p[i] = v;
}

// ------------------------------ WMMA GEMM ----------------------------------
// Y[M,256] = X[M,K] @ W[K,256].  One wave computes a 16x64 output strip:
// 4 accumulators share one A fragment per K-step (4x A-traffic reuse).
// A frag (16x4 f32): lane<16 holds {A[m,k+0],A[m,k+1]}, lane>=16 {k+2,k+3}.
// B frag (4x16 f32): mirrored: lane<16 holds {B[k+0,n],B[k+1,n]}, etc.
// C/D (16x16 f32): VGPR r -> row r (lanes 0-15) / row r+8 (lanes 16-31).
__global__ __launch_bounds__(128)
void k_gemm_wmma(const float* __restrict__ X, const float* __restrict__ W,
                 float* __restrict__ Y, int M, int K) {
  const int NC = GAT_D1;
  int wid  = (int)((blockIdx.x * blockDim.x + threadIdx.x) >> 5);
  int lane = threadIdx.x & 31;
  int mTiles = M >> 4;
  int tileM = wid >> 2;   // 4 column blocks of 64
  int nb    = wid & 3;
  if (tileM >= mTiles) return;                 // wave-uniform guard

  int half = lane >> 4;                        // 0 or 1
  int l16  = lane & 15;
  int row  = (tileM << 4) + l16;
  int kk   = half << 1;                        // 0 or 2
  int colB = (nb << 6) + l16;

  v8f acc0 = {}, acc1 = {}, acc2 = {}, acc3 = {};
  for (int k = 0; k < K; k += 4) {
    v2f a = *(const v2f*)&X[(long long)row * K + k + kk];
    const float* w0 = &W[(long long)(k + kk) * NC];
    const float* w1 = &W[(long long)(k + kk + 1) * NC];
    v2f b0 = { w0[colB +  0], w1[colB +  0] };
    v2f b1 = { w0[colB + 16], w1[colB + 16] };
    v2f b2 = { w0[colB + 32], w1[colB + 32] };
    v2f b3 = { w0[colB + 48], w1[colB + 48] };
    acc0 = __builtin_amdgcn_wmma_f32_16x16x4_f32(false, a, false, b0, (short)0, acc0, false, false);
    acc1 = __builtin_amdgcn_wmma_f32_16x16x4_f32(false, a, false, b1, (short)0, acc1, false, false);
    acc2 = __builtin_amdgcn_wmma_f32_16x16x4_f32(false, a, false, b2, (short)0, acc2, false, false);
    acc3 = __builtin_amdgcn_wmma_f32_16x16x4_f32(false, a, false, b3, (short)0, acc3, false, false);
  }
  int rbase = (tileM << 4) + (half << 3);
  for (int r = 0; r < 8; ++r) {
    float* y = &Y[(long long)(rbase + r) * NC + (nb << 6) + l16];
    y[0]  = acc0[r];
    y[16] = acc1[r];
    y[32] = acc2[r];
    y[48] = acc3[r];
  }
}

// ------------------------- per-node attention logits ------------------------
__global__ void k_logits(const float* __restrict__ H,
                         const float* __restrict__ att_s,
                         const float* __restrict__ att_d,
                         float* __restrict__ a_s, float* __restrict__ a_d) {
  int t = blockIdx.x * blockDim.x + threadIdx.x;
  if (t >= GAT_N * GAT_HEADS) return;
  int n = t >> 2, h = t & 3;
  const float* hp = &H[(long long)n * GAT_D1 + h * GAT_HID];
  const float* as = &att_s[h * GAT_HID];
  const float* ad = &att_d[h * GAT_HID];
  float s = 0.f, d = 0.f;
#pragma unroll 8
  for (int c = 0; c < GAT_HID; ++c) { float v = hp[c]; s += v * as[c]; d += v * ad[c]; }
  a_s[t] = s; a_d[t] = d;
}

// ------------------------------ edge helpers --------------------------------
__device__ __forceinline__ void edge_sd(const long long* __restrict__ ei,
                                        long long e, int& s, int& d) {
  if (e < GAT_E) { s = (int)ei[e]; d = (int)ei[GAT_E + e]; }
  else           { s = d = (int)(e - GAT_E); }
}

__device__ __forceinline__ float leaky(float x) {
  return x > 0.f ? x : NEG_SLOPE * x;
}

// pass 1: segment max of leaky-relu logits, per head
__global__ void k_edge_max(const long long* __restrict__ ei,
                           const float* __restrict__ a_s,
                           const float* __restrict__ a_d,
                           float* __restrict__ m) {
  long long e = blockIdx.x * (long long)blockDim.x + threadIdx.x;
  if (e >= GAT_TOT) return;
  int s, d; edge_sd(ei, e, s, d);
  float4 as = *(const float4*)&a_s[s * 4];
  float4 ad = *(const float4*)&a_d[d * 4];
  atomicMax(&m[d * 4 + 0], leaky(as.x + ad.x));
  atomicMax(&m[d * 4 + 1], leaky(as.y + ad.y));
  atomicMax(&m[d * 4 + 2], leaky(as.z + ad.z));
  atomicMax(&m[d * 4 + 3], leaky(as.w + ad.w));
}

// pass 2: segment sum of exp(e - m)
__global__ void k_edge_sum(const long long* __restrict__ ei,
                           const float* __restrict__ a_s,
                           const float* __restrict__ a_d,
                           const float* __restrict__ m,
                           float* __restrict__ ssum) {
  long long e = blockIdx.x * (long long)blockDim.x + threadIdx.x;
  if (e >= GAT_TOT) return;
  int s, d; edge_sd(ei, e, s, d);
  float4 as = *(const float4*)&a_s[s * 4];
  float4 ad = *(const float4*)&a_d[d * 4];
  float4 mm = *(const float4*)&m[d * 4];
  atomicAdd(&ssum[d * 4 + 0], __expf(leaky(as.x + ad.x) - mm.x));
  atomicAdd(&ssum[d * 4 + 1], __expf(leaky(as.y + ad.y) - mm.y));
  atomicAdd(&ssum[d * 4 + 2], __expf(leaky(as.z + ad.z) - mm.z));
  atomicAdd(&ssum[d * 4 + 3], __expf(leaky(as.w + ad.w) - mm.w));
}

// pass 3: out[dst] += alpha * h[src]; one wave32 per edge, 8 floats per lane
__global__ __launch_bounds__(256)
void k_edge_agg(const long long* __restrict__ ei,
                const float* __restrict__ a_s,
                const float* __restrict__ a_d,
                const float* __restrict__ m,
                const float* __restrict__ ssum,
                const float* __restrict__ H,
                float* __restrict__ out) {
  long long wid = (blockIdx.x * (long long)blockDim.x + threadIdx.x) >> 5;
  int lane = threadIdx.x & 31;
  if (wid >= GAT_TOT) return;                  // wave-uniform
  int s, d; edge_sd(ei, wid, s, d);
  int head = lane >> 3;                        // 8 lanes per head
  float e = leaky(a_s[s * 4 + head] + a_d[d * 4 + head]);
  float alpha = __expf(e - m[d * 4 + head]) / ssum[d * 4 + head];
  int col = lane << 3;                         // 8 consecutive cols per lane
  const float4 h0 = *(const float4*)&H[(long long)s * GAT_D1 + col];
  const float4 h1 = *(const float4*)&H[(long long)s * GAT_D1 + col + 4];
  float* o = &out[(long long)d * GAT_D1 + col];
  atomicAdd(&o[0], alpha * h0.x);
  atomicAdd(&o[1], alpha * h0.y);
  atomicAdd(&o[2], alpha * h0.z);
  atomicAdd(&o[3], alpha * h0.w);
  atomicAdd(&o[4], alpha * h1.x);
  atomicAdd(&o[5], alpha * h1.y);
  atomicAdd(&o[6], alpha * h1.z);
  atomicAdd(&o[7], alpha * h1.w);
}

// bias + ELU, in place
__global__ void k_bias_elu(float* __restrict__ h, const float* __restrict__ b) {
  long long i = blockIdx.x * (long long)blockDim.x + threadIdx.x;
  if (i >= (long long)GAT_N * GAT_D1) return;
  float v = h[i] + b[(int)(i & (GAT_D1 - 1))];
  h[i] = v > 0.f ? v : (__expf(v) - 1.f);
}

// global mean pool: sums + counts
__global__ void k_pool_sum(const float* __restrict__ h,
                           const long long* __restrict__ batch,
                           float* __restrict__ pooled) {
  long long i = blockIdx.x * (long long)blockDim.x + threadIdx.x;
  if (i >= (long long)GAT_N * GAT_D1) return;
  int n = (int)(i >> 8), c = (int)(i & 255);
  atomicAdd(&pooled[(int)batch[n] * GAT_D1 + c], h[i]);
}
__global__ void k_pool_cnt(const long long* __restrict__ batch,
                           float* __restrict__ cnt) {
  int n = blockIdx.x * blockDim.x + threadIdx.x;
  if (n >= GAT_N) return;
  atomicAdd(&cnt[(int)batch[n]], 1.f);
}

// final linear: out[g,o] = (pooled[g,:]/cnt[g]) @ fc_w + fc_b
__global__ void k_fc(const float* __restrict__ pooled,
                     const float* __restrict__ cnt,
                     const float* __restrict__ fw,
                     const float* __restrict__ fb,
                     float* __restrict__ out) {
  int t = blockIdx.x * blockDim.x + threadIdx.x;
  if (t >= GAT_NG * GAT_OUT) return;
  int g = t >> 6, o = t & 63;
  float inv = 1.f / fmaxf(cnt[g], 1.f);
  float acc = fb[o];
#pragma unroll 8
  for (int c = 0; c < GAT_D1; ++c)
    acc += pooled[g * GAT_D1 + c] * fw[c * GAT_OUT + o];
  out[t] = acc * 1.f, out[t] = fb[o] + (acc - fb[o]) * inv; // scale sums only
}

// ---------------------------------------------------------------------------
extern "C" void kernel_launch(void* const* d_in, const int* in_sizes, int n_in,
                              void* d_out, int out_size, void* d_ws, size_t ws_size,
                              hipStream_t stream) {
  (void)in_sizes; (void)n_in; (void)out_size; (void)ws_size;
  const float*     x        = (const float*)d_in[0];
  const long long* ei       = (const long long*)d_in[1];
  const long long* batch    = (const long long*)d_in[2];
  const float*     W1       = (const float*)d_in[3];
  const float*     att_s1   = (const float*)d_in[4];
  const float*     att_d1   = (const float*)d_in[5];
  const float*     b1       = (const float*)d_in[6];
  const float*     W2       = (const float*)d_in[7];
  const float*     att_s2   = (const float*)d_in[8];
  const float*     att_d2   = (const float*)d_in[9];
  const float*     b2       = (const float*)d_in[10];
  const float*     fc_w     = (const float*)d_in[11];
  const float*     fc_b     = (const float*)d_in[12];
  float*           out      = (float*)d_out;

  // workspace carve-up (all 16B aligned)
  char* ws = (char*)d_ws;
  const size_t featB = (size_t)GAT_N * GAT_D1 * sizeof(float);   // 51.2 MB
  float* bufA   = (float*)(ws);                       // transformed features
  float* bufB   = (float*)(ws + featB);               // aggregation target
  float* a_s    = (float*)(ws + 2 * featB);
  float* a_d    = (float*)(ws + 2 * featB +  800000);
  float* mx     = (float*)(ws + 2 * featB + 1600000);
  float* sm     = (float*)(ws + 2 * featB + 2400000);
  float* pooled = (float*)(ws + 2 * featB + 3200000);
  float* cnt    = (float*)(ws + 2 * featB + 3200000 + GAT_NG * GAT_D1 * 4);

  const int NH   = GAT_N * GAT_HEADS;                    // 200000
  const int FEAT = GAT_N * GAT_D1;                       // 12.8M
  dim3 blk(256);
  dim3 gFill((FEAT + 255) / 256);
  dim3 gNH((NH + 255) / 256);
  dim3 gEdge((unsigned)((GAT_TOT + 255) / 256));
  dim3 gAgg((unsigned)(GAT_TOT / 8));                    // 1 wave / edge, 8 waves / block
  dim3 gGemm(GAT_N / 16);                                // 4 waves x (16x64) per block

  // ---------------- layer 1 ----------------
  k_gemm_wmma<<<gGemm, 128, 0, stream>>>(x, W1, bufA, GAT_N, GAT_FIN);
  k_logits<<<gNH, blk, 0, stream>>>(bufA, att_s1, att_d1, a_s, a_d);
  k_fill<<<(NH + 255) / 256, blk, 0, stream>>>(mx, -1e30f, NH);
  k_fill<<<(NH + 255) / 256, blk, 0, stream>>>(sm, 0.f, NH);
  k_edge_max<<<gEdge, blk, 0, stream>>>(ei, a_s, a_d, mx);
  k_edge_sum<<<gEdge, blk, 0, stream>>>(ei, a_s, a_d, mx, sm);
  k_fill<<<gFill, blk, 0, stream>>>(bufB, 0.f, FEAT);
  k_edge_agg<<<gAgg, blk, 0, stream>>>(ei, a_s, a_d, mx, sm, bufA, bufB);
  k_bias_elu<<<gFill, blk, 0, stream>>>(bufB, b1);

  // ---------------- layer 2 ----------------
  k_gemm_wmma<<<gGemm, 128, 0, stream>>>(bufB, W2, bufA, GAT_N, GAT_D1);
  k_logits<<<gNH, blk, 0, stream>>>(bufA, att_s2, att_d2, a_s, a_d);
  k_fill<<<(NH + 255) / 256, blk, 0, stream>>>(mx, -1e30f, NH);
  k_fill<<<(NH + 255) / 256, blk, 0, stream>>>(sm, 0.f, NH);
  k_edge_max<<<gEdge, blk, 0, stream>>>(ei, a_s, a_d, mx);
  k_edge_sum<<<gEdge, blk, 0, stream>>>(ei, a_s, a_d, mx, sm);
  k_fill<<<gFill, blk, 0, stream>>>(bufB, 0.f, FEAT);
  k_edge_agg<<<gAgg, blk, 0, stream>>>(ei, a_s, a_d, mx, sm, bufA, bufB);
  k_bias_elu<<<gFill, blk, 0, stream>>>(bufB, b2);

  // ---------------- pool + fc ----------------
  k_fill<<<(GAT_NG * GAT_D1 + 255) / 256, blk, 0, stream>>>(pooled, 0.f, GAT_NG * GAT_D1);
  k_fill<<<(GAT_NG + 255) / 256, blk, 0, stream>>>(cnt, 0.f, GAT_NG);
  k_pool_sum<<<gFill, blk, 0, stream>>>(bufB, batch, pooled);
  k_pool_cnt<<<(GAT_N + 255) / 256, blk, 0, stream>>>(batch, cnt);
  k_fc<<<(GAT_NG * GAT_OUT + 255) / 256, blk, 0, stream>>>(pooled, cnt, fc_w, fc_b, out);
}